// EpsSupInfoNCELoss_32684701123305
// MI455X (gfx1250) — compile-verified
//
#include <hip/hip_runtime.h>

typedef float v8f __attribute__((ext_vector_type(8)));
typedef float v2f __attribute__((ext_vector_type(2)));
typedef int   v4i __attribute__((ext_vector_type(4)));

// Address-space-qualified pointer typedefs so the AS attribute binds properly.
typedef __attribute__((address_space(1))) v4i* gptr_v4i;   // global
typedef __attribute__((address_space(3))) v4i* lptr_v4i;   // LDS

#define BN       8192
#define DK       128
#define NCLS     100
#define INV_T    14.285714285714285714f   /* 1/0.07 */
#define MAXL     14.285714285714285714f   /* max logit: cosine<=1 -> 1/T */
#define EPS_     0.25f
#define THREADS  256
#define ROWS_PB  128                      /* 8 waves * 16 rows */
#define JCHUNK   32
#define LSTRIDE  132                      /* padded LDS row stride (floats) */

#if __has_builtin(__builtin_amdgcn_global_load_async_to_lds_b128)
#define HAVE_ASYNC_LDS 1
#else
#define HAVE_ASYNC_LDS 0
#endif

__device__ __forceinline__ void async_wait0() {
#if __has_builtin(__builtin_amdgcn_s_wait_asynccnt)
  __builtin_amdgcn_s_wait_asynccnt(0);
#else
  asm volatile("s_wait_asynccnt 0x0" ::: "memory");
#endif
}

// ---------------------------------------------------------------------------
// Kernel 0: label histogram -> invcnt[j] = 1 / (count(label_j) - 1)
// ---------------------------------------------------------------------------
__global__ __launch_bounds__(256) void k_hist(const int* __restrict__ labels,
                                              float* __restrict__ invcnt) {
  __shared__ int hist[NCLS];
  const int t = threadIdx.x;
  if (t < NCLS) hist[t] = 0;
  __syncthreads();
  for (int j = t; j < BN; j += blockDim.x) atomicAdd(&hist[labels[j]], 1);
  __syncthreads();
  for (int j = t; j < BN; j += blockDim.x) {
    const int c = hist[labels[j]] - 1;
    invcnt[j] = (c > 0) ? (1.0f / (float)c) : 0.0f;
  }
}

// ---------------------------------------------------------------------------
// Fused GEMM passes. PHASE 0: base[i] = logsumexp_{label_j != label_i} l_ij
// PHASE 1: partial[blk] = sum over same-label pairs of
//          (logaddexp(l - EPS, base_i) - l) * invcnt[j] / n
// ---------------------------------------------------------------------------
template <int PHASE>
__global__ __launch_bounds__(THREADS) void k_gemm(const float* __restrict__ E,
                                                  const int* __restrict__ labels,
                                                  const float* __restrict__ invcnt,
                                                  float* __restrict__ base,
                                                  float* __restrict__ partial) {
  __shared__ float bufB[2][JCHUNK * LSTRIDE];
  __shared__ float red[THREADS];

  const int t    = threadIdx.x;
  const int wave = t >> 5;
  const int lane = t & 31;
  const int half = lane >> 4;   // 0: lanes 0-15, 1: lanes 16-31
  const int lid  = lane & 15;

  const int rowBase = blockIdx.x * ROWS_PB + wave * 16;

  // ---- A panel, resident in VGPRs for the whole N loop -------------------
  // A fragment layout (16x4 f32): lanes 0-15 hold row M=lid, K = k0+{0,1};
  // lanes 16-31 hold row M=lid, K = k0+{2,3}.  32 fragments cover K=128.
  v2f af[32];
  const float* arow = E + (size_t)(rowBase + lid) * DK + 2 * half;
#pragma unroll
  for (int f = 0; f < 32; ++f) af[f] = *(const v2f*)(arow + 4 * f);

  // Per-lane row metadata for C/D layout: acc[v] is row (v + 8*half).
  int   li[8];
  float bi[8];
#pragma unroll
  for (int v = 0; v < 8; ++v) {
    const int r = rowBase + v + 8 * half;
    li[v] = labels[r];
    bi[v] = (PHASE == 1) ? base[r] : 0.0f;
  }

  float s[8];
#pragma unroll
  for (int v = 0; v < 8; ++v) s[v] = 0.0f;
  float contrib = 0.0f;

  // ---- staging: one 32-row x 128-col panel of E into padded LDS ----------
  auto stage_async = [&](int J0, int bsel) {
#if HAVE_ASYNC_LDS
#pragma unroll
    for (int k = 0; k < 4; ++k) {
      const int f4 = k * THREADS + t;   // 1024 float4s total
      const int r  = f4 >> 5;           // 32 float4 per row
      const int c4 = f4 & 31;
      const float* gsrc = E + (size_t)(J0 + r) * DK + c4 * 4;
      float*       ldst = &bufB[bsel][r * LSTRIDE + c4 * 4];
      __builtin_amdgcn_global_load_async_to_lds_b128(
          (gptr_v4i)gsrc, (lptr_v4i)ldst, 0, 0);
    }
#else
    (void)J0; (void)bsel;
#endif
  };
  auto stage_sync = [&](int J0, int bsel) {
#pragma unroll
    for (int k = 0; k < 4; ++k) {
      const int f4 = k * THREADS + t;
      const int r  = f4 >> 5;
      const int c4 = f4 & 31;
      const float4 val = *(const float4*)(E + (size_t)(J0 + r) * DK + c4 * 4);
      *(float4*)(&bufB[bsel][r * LSTRIDE + c4 * 4]) = val;
    }
  };

  // ---- one 32-column chunk: two 16x16 WMMA tiles per wave ----------------
  auto compute = [&](int J0, int bsel) {
#pragma unroll
    for (int jt = 0; jt < 2; ++jt) {
      const int jcol = J0 + jt * 16 + lid;   // this lane's output column
      const int lj   = labels[jcol];

      const float* brow = &bufB[bsel][(jt * 16 + lid) * LSTRIDE + 2 * half];
      v8f acc = {};
#pragma unroll
      for (int f = 0; f < 32; ++f) {
        const v2f b = *(const v2f*)(brow + 4 * f);
        acc = __builtin_amdgcn_wmma_f32_16x16x4_f32(
            /*neg_a=*/false, af[f], /*neg_b=*/false, b,
            /*c_mod=*/(short)0, acc, /*reuse_a=*/false, /*reuse_b=*/false);
      }

      if (PHASE == 0) {
#pragma unroll
        for (int v = 0; v < 8; ++v) {
          const float l = acc[v] * INV_T;
          if (li[v] != lj) s[v] += __expf(l - MAXL);
        }
      } else {
        const float ic = invcnt[jcol];
#pragma unroll
        for (int v = 0; v < 8; ++v) {
          const int r = rowBase + v + 8 * half;
          if (li[v] == lj && r != jcol) {
            const float l  = acc[v] * INV_T;
            const float a  = l - EPS_;
            const float b2 = bi[v];
            const float m2 = fmaxf(a, b2);
            const float lad = m2 + __logf(__expf(a - m2) + __expf(b2 - m2));
            contrib += (lad - l) * ic;
          }
        }
      }
    }
  };

  const int NCHUNK = BN / JCHUNK;
#if HAVE_ASYNC_LDS
  // Double-buffered pipeline: copy of chunk c+1 (ASYNCcnt, no VGPRs)
  // overlaps the 64 chained WMMAs of chunk c. One barrier per chunk.
  stage_async(0, 0);
  for (int c = 0; c < NCHUNK; ++c) {
    async_wait0();
    __syncthreads();
    if (c + 1 < NCHUNK) stage_async((c + 1) * JCHUNK, (c + 1) & 1);
    compute(c * JCHUNK, c & 1);
  }
#else
  for (int c = 0; c < NCHUNK; ++c) {
    __syncthreads();
    stage_sync(c * JCHUNK, 0);
    __syncthreads();
    compute(c * JCHUNK, 0);
  }
#endif

  if (PHASE == 0) {
    // Reduce each row-sum across the 16 lanes of its half-wave.
#pragma unroll
    for (int v = 0; v < 8; ++v) {
      float x = s[v];
#pragma unroll
      for (int m = 1; m < 16; m <<= 1) x += __shfl_xor(x, m, 32);
      s[v] = x;
    }
    if (lid == 0) {
#pragma unroll
      for (int v = 0; v < 8; ++v) {
        const int r = rowBase + v + 8 * half;
        base[r] = MAXL + __logf(s[v]);
      }
    }
  } else {
    // Deterministic in-block tree reduction.
    red[t] = contrib;
    __syncthreads();
    for (int off = THREADS / 2; off > 0; off >>= 1) {
      if (t < off) red[t] += red[t + off];
      __syncthreads();
    }
    if (t == 0) partial[blockIdx.x] = red[0] * (1.0f / (float)BN);
  }
}

// ---------------------------------------------------------------------------
// Finalize: fixed-order sum of per-block partials (deterministic).
// ---------------------------------------------------------------------------
__global__ void k_final(const float* __restrict__ partial,
                        float* __restrict__ out, int nblk) {
  if (threadIdx.x == 0) {
    float s = 0.0f;
    for (int i = 0; i < nblk; ++i) s += partial[i];
    out[0] = s;
  }
}

// ---------------------------------------------------------------------------
extern "C" void kernel_launch(void* const* d_in, const int* in_sizes, int n_in,
                              void* d_out, int out_size, void* d_ws, size_t ws_size,
                              hipStream_t stream) {
  (void)in_sizes; (void)n_in; (void)out_size; (void)ws_size;
  const float* embeds = (const float*)d_in[0];
  const int*   labels = (const int*)d_in[1];

  float* ws      = (float*)d_ws;
  float* base    = ws;             // 8192 floats
  float* invcnt  = ws + BN;        // 8192 floats
  float* partial = ws + 2 * BN;    // 64 floats

  const int nblk = BN / ROWS_PB;   // 64

  k_hist<<<1, 256, 0, stream>>>(labels, invcnt);
  k_gemm<0><<<nblk, THREADS, 0, stream>>>(embeds, labels, invcnt, base, partial);
  k_gemm<1><<<nblk, THREADS, 0, stream>>>(embeds, labels, invcnt, base, partial);
  k_final<<<1, 32, 0, stream>>>(partial, (float*)d_out, nblk);
}